// VQEmbeddingEMA_86560770884062
// MI455X (gfx1250) — compile-verified
//
#include <hip/hip_runtime.h>
#include <hip/hip_bf16.h>
#include <math.h>

// ---------------------------------------------------------------------------
// VQ-VAE codebook quantization for MI455X (gfx1250, wave32, WMMA).
//   x:         [131072, 64] f32 (flattened from [32,64,64,64])
//   embedding: [512, 64]    f32
//   out:       quantized [131072*64] f32, then loss, then perplexity
// Memory-bound (~67 MB @ 23.3 TB/s ≈ 2.9 us); distances via V_WMMA_F32_16X16X4_F32.
// ---------------------------------------------------------------------------

#define NROWS   131072      // number of x vectors
#define DDIM    64
#define MCODES  512
#define XS_STRIDE 68        // padded LDS row stride (floats): 272B -> bank-conflict free

typedef __attribute__((ext_vector_type(2))) float v2f;
typedef __attribute__((ext_vector_type(8))) float v8f;

// workspace layout (floats): [0..511]=enorm, [512..1023]=counts, [1024]=loss_acc
#define WS_ENORM  0
#define WS_COUNTS 512
#define WS_LOSS   1024

// ---------------------------------------------------------------------------
// Kernel 0: enorm precompute + per-launch workspace re-init (graph-safe).
// ---------------------------------------------------------------------------
__global__ void vq_prep(const float* __restrict__ emb, float* __restrict__ ws) {
    int t = threadIdx.x;                 // 0..511, one code per thread
    const float* e = emb + t * DDIM;
    float s = 0.0f;
    #pragma unroll
    for (int d = 0; d < DDIM; ++d) { float v = e[d]; s += v * v; }
    ws[WS_ENORM + t]  = s;
    ws[WS_COUNTS + t] = 0.0f;
    if (t == 0) ws[WS_LOSS] = 0.0f;
}

// ---------------------------------------------------------------------------
// Kernel 1: per-16-row tile: WMMA distance scores, argmin, gather, partials.
// 256 threads = 8 waves; wave w covers codes [64w, 64w+64) as 4 x 16-col tiles.
// ---------------------------------------------------------------------------
__global__ void __launch_bounds__(256) vq_main(const float* __restrict__ x,
                                               const float* __restrict__ emb,
                                               float* __restrict__ ws,
                                               float* __restrict__ out) {
    __shared__ float xs[16 * XS_STRIDE];      // padded 16x64 x-tile
    __shared__ float wb_score[8 * 16];        // per-wave best score per row
    __shared__ int   wb_idx[8 * 16];
    __shared__ int   rowidx[16];

    const int t       = threadIdx.x;
    const int lane    = t & 31;
    const int wave    = t >> 5;
    const int m       = lane & 15;            // row within tile (A) / col (B)
    const int khalf   = lane >> 4;            // selects K pair {0,1} vs {2,3}
    const int rowBase = blockIdx.x * 16;

    // ---- stage x tile: coalesced b128 load, padded LDS store -------------
    {
        const int r  = t >> 4;                // 0..15
        const int c4 = (t & 15) * 4;          // 0..60 step 4
        const float4 xv = *(const float4*)(x + (size_t)rowBase * DDIM + 4 * t);
        *(float4*)(&xs[r * XS_STRIDE + c4]) = xv;
    }
    __syncthreads();

    // ---- WMMA score accumulation + per-lane argmin -----------------------
    float bs[8]; int bi[8];
    #pragma unroll
    for (int r = 0; r < 8; ++r) { bs[r] = 3.402823466e38f; bi[r] = 0x7fffffff; }

    const float* enorm = ws + WS_ENORM;

    #pragma unroll
    for (int ct = 0; ct < 4; ++ct) {
        const int colBase = wave * 64 + ct * 16;
        const int col     = colBase + m;      // this lane's code column
        const float en    = enorm[col];

        v8f acc = {0.f, 0.f, 0.f, 0.f, 0.f, 0.f, 0.f, 0.f};
        #pragma unroll
        for (int kk = 0; kk < 16; ++kk) {
            const int k0 = kk * 4 + 2 * khalf;
            // A 16x4 f32: lane m=row, VGPRv = x[m][k0+v]   (ds_load_b64)
            v2f a = *(const v2f*)(&xs[m * XS_STRIDE + k0]);
            // B 4x16 f32: lane m=col, VGPRv = E[col][k0+v] (global_load_b64, L2 hot)
            v2f b = *(const v2f*)(emb + (size_t)col * DDIM + k0);
            acc = __builtin_amdgcn_wmma_f32_16x16x4_f32(
                /*neg_a=*/false, a, /*neg_b=*/false, b,
                /*c_mod=*/(short)0, acc, /*reuse_a=*/false, /*reuse_b=*/false);
        }
        // score = enorm[col] - 2*dot  (xnorm constant per row -> dropped)
        #pragma unroll
        for (int r = 0; r < 8; ++r) {
            float sc = fmaf(-2.0f, acc[r], en);
            if (sc < bs[r] || (sc == bs[r] && col < bi[r])) { bs[r] = sc; bi[r] = col; }
        }
    }

    // ---- butterfly argmin across each 16-lane half (stays within half) ---
    #pragma unroll
    for (int mask = 1; mask <= 8; mask <<= 1) {
        #pragma unroll
        for (int r = 0; r < 8; ++r) {
            float os = __shfl_xor(bs[r], mask, 32);
            int   oi = __shfl_xor(bi[r], mask, 32);
            if (os < bs[r] || (os == bs[r] && oi < bi[r])) { bs[r] = os; bi[r] = oi; }
        }
    }
    if (m == 0) {                             // lanes 0 and 16 publish
        #pragma unroll
        for (int r = 0; r < 8; ++r) {
            const int row = r + 8 * khalf;    // ISA C/D layout: VGPR r -> row r / r+8
            wb_score[wave * 16 + row] = bs[r];
            wb_idx[wave * 16 + row]   = bi[r];
        }
    }
    __syncthreads();

    // ---- cross-wave argmin (first-index tie-break, like jnp.argmin) ------
    if (t < 16) {
        float best = 3.402823466e38f; int bidx = 0x7fffffff;
        #pragma unroll
        for (int w = 0; w < 8; ++w) {
            float s = wb_score[w * 16 + t];
            int   i = wb_idx[w * 16 + t];
            if (s < best || (s == best && i < bidx)) { best = s; bidx = i; }
        }
        rowidx[t] = bidx;
    }
    __syncthreads();

    // ---- output: gather code, coalesced b128 store, loss/count partials --
    {
        const int r    = t >> 4;
        const int c4   = (t & 15) * 4;
        const int code = rowidx[r];
        const float4 e  = *(const float4*)(emb + (size_t)code * DDIM + c4);
        const float4 xv = *(const float4*)(&xs[r * XS_STRIDE + c4]);
        // quantized_st forward value == embedding[idx]
        *(float4*)(out + (size_t)rowBase * DDIM + 4 * t) = e;

        float dx = xv.x - e.x, dy = xv.y - e.y, dz = xv.z - e.z, dw = xv.w - e.w;
        float err = dx * dx + dy * dy + dz * dz + dw * dw;
        #pragma unroll
        for (int mask = 16; mask >= 1; mask >>= 1)
            err += __shfl_xor(err, mask, 32);
        if (lane == 0) atomicAdd(ws + WS_LOSS, err);
        if ((t & 15) == 0) atomicAdd(ws + WS_COUNTS + code, 1.0f);
    }
}

// ---------------------------------------------------------------------------
// Kernel 2: finalize loss + perplexity.
// ---------------------------------------------------------------------------
__global__ void vq_fin(const float* __restrict__ ws, float* __restrict__ out) {
    __shared__ float red[512];
    int t = threadIdx.x;                      // 0..511
    float c = ws[WS_COUNTS + t];
    float p = c * (1.0f / (float)NROWS);
    red[t] = p * logf(p + 1e-10f);
    __syncthreads();
    #pragma unroll
    for (int s = 256; s > 0; s >>= 1) {
        if (t < s) red[t] += red[t + s];
        __syncthreads();
    }
    if (t == 0) {
        const size_t qn = (size_t)NROWS * DDIM;
        out[qn]     = 0.25f * ws[WS_LOSS] / (float)qn;   // commitment loss
        out[qn + 1] = expf(-red[0]);                     // perplexity
    }
}

// ---------------------------------------------------------------------------
extern "C" void kernel_launch(void* const* d_in, const int* in_sizes, int n_in,
                              void* d_out, int out_size, void* d_ws, size_t ws_size,
                              hipStream_t stream) {
    const float* x   = (const float*)d_in[0];   // [131072, 64]
    const float* emb = (const float*)d_in[1];   // [512, 64]
    float* out = (float*)d_out;
    float* ws  = (float*)d_ws;

    vq_prep<<<1, MCODES, 0, stream>>>(emb, ws);
    vq_main<<<NROWS / 16, 256, 0, stream>>>(x, emb, ws, out);
    vq_fin <<<1, MCODES, 0, stream>>>(ws, out);
}